// GSSM_v1_35605278884492
// MI455X (gfx1250) — compile-verified
//
#include <hip/hip_runtime.h>
#include <hip/hip_bf16.h>

// GSSM (Mamba-like selective scan over curve-ordered tokens) for gfx1250.
// All GEMMs use V_WMMA_F32_16X16X4_F32 (full f32 precision on the matrix pipe;
// workload is latency-bound by the L=8192 recurrence, not matrix throughput).

typedef __attribute__((ext_vector_type(2))) float v2f;
typedef __attribute__((ext_vector_type(8))) float v8f;

#define NB 4
#define LSEQ 8192
#define NTOK (NB * LSEQ)          // 32768
#define DIMF 128
#define DIN 256
#define DST 16
#define DCONV 4
#define DTR 8

// ---------------------------------------------------------------------------
// f32 WMMA 16x16 tile: C += A(16xK, row-major lda=K) * W(col-major via rows of
// weight matrix: B[k][n] = Wrow_n[k], Wrow_n = BT + n*ldb).
// A 16x4 f32 layout: lane l: m = l&15, holds K = 2*(l>>4)+{0,1} in {a.x,a.y}.
// B 4x16  f32 layout: lane l: n = l&15, holds K = 2*(l>>4)+{0,1} in {b.x,b.y}.
// C/D 16x16: VGPR r: row M = r + 8*(l>>4), col N = l&15.
// ---------------------------------------------------------------------------
__device__ __forceinline__ v8f wmma_tile_f32(const float* __restrict__ A, int lda,
                                             const float* __restrict__ BT, int ldb,
                                             int K, v8f acc) {
  const int lane = threadIdx.x & 31;
  const int m    = lane & 15;
  const int kh   = (lane >> 4) << 1;
  for (int k = 0; k < K; k += 4) {
    v2f a, b;
    a.x = A[m * lda + k + kh];
    a.y = A[m * lda + k + kh + 1];
    b.x = BT[m * ldb + k + kh];
    b.y = BT[m * ldb + k + kh + 1];
    acc = __builtin_amdgcn_wmma_f32_16x16x4_f32(false, a, false, b,
                                                (short)0, acc, false, false);
  }
  return acc;
}

// ---------------------------------------------------------------------------
// K1: window_pos features + curve ranks
// ---------------------------------------------------------------------------
__global__ void k_pos(const int* __restrict__ coords, const int* __restrict__ curve,
                      float* __restrict__ pos, int* __restrict__ ranks) {
  int n = blockIdx.x * 256 + threadIdx.x;
  if (n >= NTOK) return;
  int zc = coords[n * 4 + 1];
  int yc = coords[n * 4 + 2];
  int xc = coords[n * 4 + 3];
  float yf = (float)yc, xf = (float)xc;
  float ycnt = 0.f, xcnt = 0.f, wy = 0.f, wx = 0.f, ly = 0.f, lx = 0.f;
#pragma unroll
  for (int w = 0; w < 11; ++w) {
    float s = 48.f * (float)w;
    float yin = (yf >= s && yf < s + 64.f) ? 1.f : 0.f;
    float xin = (xf >= s && xf < s + 64.f) ? 1.f : 0.f;
    float wi = (float)w * (1.f / 11.f);
    ycnt += yin; xcnt += xin;
    wy += yin * wi; wx += xin * wi;
    ly += yin * (yf - s) * (1.f / 64.f);
    lx += xin * (xf - s) * (1.f / 64.f);
  }
  pos[n * 5 + 0] = (float)zc * (1.f / 32.f);
  pos[n * 5 + 1] = wx * ycnt;
  pos[n * 5 + 2] = wy * xcnt;
  pos[n * 5 + 3] = lx * ycnt;
  pos[n * 5 + 4] = ly * xcnt;
  long flat = (long)zc * (512L * 512L) + (long)yc * 512L + (long)xc;
  ranks[n] = curve[flat];
}

// ---------------------------------------------------------------------------
// K2: pe = relu(pos@W1^T+b1)@W2^T+b2 ; f = feats + pe   (W2 layer via WMMA)
// 64 tokens / block, 256 threads (8 waves), h1 staged in LDS (32 KB)
// ---------------------------------------------------------------------------
__global__ void k_pe(const float* __restrict__ pos, const float* __restrict__ w1,
                     const float* __restrict__ b1, const float* __restrict__ w2,
                     const float* __restrict__ b2, const float* __restrict__ feats,
                     float* __restrict__ fout) {
  __shared__ float h1[64 * DIMF];
  const int t0 = blockIdx.x * 64;
  for (int i = threadIdx.x; i < 64 * DIMF; i += 256) {
    int tt = i >> 7, d = i & 127;
    const float* p = pos + (size_t)(t0 + tt) * 5;
    float a = b1[d];
#pragma unroll
    for (int j = 0; j < 5; ++j) a += p[j] * w1[d * 5 + j];
    h1[i] = fmaxf(a, 0.f);
  }
  __syncthreads();
  const int wave = threadIdx.x >> 5, lane = threadIdx.x & 31;
  for (int tile = wave; tile < 32; tile += 8) {   // 4 row-tiles x 8 col-tiles
    int tr = tile >> 3, tc = tile & 7;
    v8f acc = {};
    acc = wmma_tile_f32(h1 + tr * 16 * DIMF, DIMF,
                        w2 + (size_t)(tc * 16) * DIMF + (size_t)(lane & 15) * 0 +
                            (size_t)0, DIMF, DIMF, acc);
    // NOTE: wmma_tile_f32 indexes BT by n=lane&15 internally via m; pass base:
    // (call above already used base w2 + tc*16*DIMF and n stride DIMF)
    int col = tc * 16 + (lane & 15);
#pragma unroll
    for (int r = 0; r < 8; ++r) {
      int row = t0 + tr * 16 + r + 8 * (lane >> 4);
      fout[(size_t)row * DIMF + col] =
          feats[(size_t)row * DIMF + col] + acc[r] + b2[col];
    }
  }
}

// ---------------------------------------------------------------------------
// K3: per-batch stable argsort via bitonic sort on 64-bit (rank<<13)|idx keys
// 1 workgroup per batch, 1024 threads, 64 KB LDS
// ---------------------------------------------------------------------------
__global__ void k_sort(const int* __restrict__ ranks, int* __restrict__ perm,
                       int* __restrict__ invp) {
  __shared__ unsigned long long keys[LSEQ];
  const int b = blockIdx.x;
  for (int i = threadIdx.x; i < LSEQ; i += 1024)
    keys[i] = ((unsigned long long)(unsigned)ranks[b * LSEQ + i] << 13) |
              (unsigned long long)i;
  __syncthreads();
  for (int k = 2; k <= LSEQ; k <<= 1) {
    for (int j = k >> 1; j > 0; j >>= 1) {
      for (int i = threadIdx.x; i < LSEQ; i += 1024) {
        int ixj = i ^ j;
        if (ixj > i) {
          bool up = ((i & k) == 0);
          unsigned long long a = keys[i], c = keys[ixj];
          if (up ? (a > c) : (a < c)) { keys[i] = c; keys[ixj] = a; }
        }
      }
      __syncthreads();
    }
  }
  for (int i = threadIdx.x; i < LSEQ; i += 1024) {
    int src = (int)(keys[i] & 8191ull);
    perm[b * LSEQ + i] = src;       // perm[t] = source index (local)
    invp[b * LSEQ + src] = i;       // inverse permutation (local)
  }
}

// ---------------------------------------------------------------------------
// K4: gather by perm + RMSNorm * rms_w   (1 token per 128-thread block)
// ---------------------------------------------------------------------------
__global__ void k_rms_perm(const float* __restrict__ f, const int* __restrict__ perm,
                           const float* __restrict__ rms_w, float* __restrict__ h) {
  const int tok = blockIdx.x;
  const int b = tok >> 13;
  const int src = perm[tok];
  const float* row = f + (size_t)(b * LSEQ + src) * DIMF;
  const int d = threadIdx.x;
  float v = row[d];
  float ss = v * v;
#pragma unroll
  for (int off = 16; off > 0; off >>= 1) ss += __shfl_xor(ss, off, 32);
  __shared__ float red[4];
  if ((threadIdx.x & 31) == 0) red[threadIdx.x >> 5] = ss;
  __syncthreads();
  float total = red[0] + red[1] + red[2] + red[3];
  float inv = rsqrtf(total * (1.f / DIMF) + 1e-5f);
  h[(size_t)tok * DIMF + d] = v * inv * rms_w[d];
}

// ---------------------------------------------------------------------------
// K5/K7/K10: generic WMMA GEMM  C(Nx NCMAX) = A(N x K) * W(NCMAX x K)^T
// 64 tokens / block, A staged in LDS; NC = cols padded to multiple of 16.
// ---------------------------------------------------------------------------
template <int K, int NC, int NCMAX>
__global__ void k_gemm(const float* __restrict__ Ain, const float* __restrict__ Wt,
                       float* __restrict__ Cout) {
  __shared__ float Al[64 * K];
  const size_t t0 = (size_t)blockIdx.x * 64;
  for (int i = threadIdx.x * 4; i < 64 * K; i += 256 * 4)
    *(float4*)&Al[i] = *(const float4*)&Ain[t0 * K + i];
  __syncthreads();
  const int wave = threadIdx.x >> 5, lane = threadIdx.x & 31;
  const int ct = NC / 16;
  for (int tile = wave; tile < 4 * ct; tile += 8) {
    int tr = tile / ct, tc = tile % ct;
    int col = tc * 16 + (lane & 15);
    int colc = col < NCMAX ? col : NCMAX - 1;   // clamp OOB weight loads
    v8f acc = {};
    acc = wmma_tile_f32(Al + tr * 16 * K, K,
                        Wt + (size_t)(colc - (lane & 15)) * K, K, K, acc);
    if (col < NCMAX) {
#pragma unroll
      for (int r = 0; r < 8; ++r) {
        int row = tr * 16 + r + 8 * (lane >> 4);
        Cout[(t0 + row) * (size_t)NCMAX + col] = acc[r];
      }
    }
  }
}

// ---------------------------------------------------------------------------
// K6: causal depthwise conv (D_CONV=4) + SiLU   (thread per (token, channel))
// ---------------------------------------------------------------------------
__global__ void k_conv(const float* __restrict__ xz, const float* __restrict__ cw,
                       const float* __restrict__ cb, float* __restrict__ xcb) {
  int gid = blockIdx.x * 256 + threadIdx.x;
  int c = gid & 255;
  int idx = gid >> 8;
  if (idx >= NTOK) return;
  int t = idx & (LSEQ - 1);
  float acc = cb[c];
#pragma unroll
  for (int k = 0; k < DCONV; ++k) {
    int s = t - (DCONV - 1) + k;
    if (s >= 0)
      acc += xz[(size_t)(idx - t + s) * 512 + c] * cw[c * DCONV + k];
  }
  float sig = 1.f / (1.f + __expf(-acc));
  xcb[(size_t)idx * DIN + c] = acc * sig;
}

// ---------------------------------------------------------------------------
// K8: dt = softplus(dbl[:, :8] @ dt_proj_w^T + dt_proj_b)
// ---------------------------------------------------------------------------
__global__ void k_dt(const float* __restrict__ dbl, const float* __restrict__ dw,
                     const float* __restrict__ db, float* __restrict__ dtb) {
  int gid = blockIdx.x * 256 + threadIdx.x;
  int c = gid & 255;
  int idx = gid >> 8;
  if (idx >= NTOK) return;
  const float* dr = dbl + (size_t)idx * 40;
  float a = db[c];
#pragma unroll
  for (int r = 0; r < DTR; ++r) a += dr[r] * dw[c * DTR + r];
  float sp = (a > 20.f) ? a : log1pf(__expf(a));
  dtb[(size_t)idx * DIN + c] = sp;
}

// ---------------------------------------------------------------------------
// K9: selective scan (sequential over L), fused D-skip + SiLU(z) gate.
// 4 blocks (batches) x 256 threads (channels); 16 states in registers.
// ---------------------------------------------------------------------------
__global__ void k_scan(const float* __restrict__ dtb, const float* __restrict__ xcb,
                       const float* __restrict__ dbl, const float* __restrict__ xz,
                       const float* __restrict__ A_log, const float* __restrict__ Dp,
                       float* __restrict__ yyb) {
  const int b = blockIdx.x, c = threadIdx.x;
  float A[DST], hs[DST];
#pragma unroll
  for (int s = 0; s < DST; ++s) {
    A[s] = -__expf(A_log[c * DST + s]);
    hs[s] = 0.f;
  }
  const float dpc = Dp[c];
  for (int t = 0; t < LSEQ; ++t) {
    size_t idx = (size_t)b * LSEQ + t;
    float dtv = dtb[idx * DIN + c];
    float xv  = xcb[idx * DIN + c];
    float zv  = xz[idx * 512 + 256 + c];
    const float* dr = dbl + idx * 40;
    if (t + 1 < LSEQ) {
      __builtin_prefetch(dtb + (idx + 1) * DIN + c, 0, 1);
      __builtin_prefetch(xcb + (idx + 1) * DIN + c, 0, 1);
    }
    float y = 0.f;
#pragma unroll
    for (int s = 0; s < DST; ++s) {
      float dA = __expf(dtv * A[s]);
      hs[s] = dA * hs[s] + dtv * dr[8 + s] * xv;
      y += hs[s] * dr[24 + s];
    }
    float sig = 1.f / (1.f + __expf(-zv));
    yyb[idx * DIN + c] = (y + dpc * xv) * (zv * sig);
  }
}

// ---------------------------------------------------------------------------
// K11: inverse permute + LayerNorm -> d_out
// ---------------------------------------------------------------------------
__global__ void k_ln(const float* __restrict__ obuf, const int* __restrict__ invp,
                     const float* __restrict__ w, const float* __restrict__ bias,
                     float* __restrict__ out) {
  const int tok = blockIdx.x;
  const int b = tok >> 13;
  const int src = invp[tok];
  const float* row = obuf + (size_t)(b * LSEQ + src) * DIMF;
  const int d = threadIdx.x;
  float v = row[d];
  float s = v, s2 = v * v;
#pragma unroll
  for (int off = 16; off > 0; off >>= 1) {
    s  += __shfl_xor(s, off, 32);
    s2 += __shfl_xor(s2, off, 32);
  }
  __shared__ float rs[4], rs2[4];
  if ((threadIdx.x & 31) == 0) { rs[threadIdx.x >> 5] = s; rs2[threadIdx.x >> 5] = s2; }
  __syncthreads();
  float S  = rs[0] + rs[1] + rs[2] + rs[3];
  float S2 = rs2[0] + rs2[1] + rs2[2] + rs2[3];
  float mu = S * (1.f / DIMF);
  float var = S2 * (1.f / DIMF) - mu * mu;
  out[(size_t)tok * DIMF + d] = (v - mu) * rsqrtf(var + 1e-5f) * w[d] + bias[d];
}

// ---------------------------------------------------------------------------
extern "C" void kernel_launch(void* const* d_in, const int* in_sizes, int n_in,
                              void* d_out, int out_size, void* d_ws, size_t ws_size,
                              hipStream_t stream) {
  const float* feats     = (const float*)d_in[0];
  const int*   coords    = (const int*)d_in[1];
  const int*   curve     = (const int*)d_in[2];
  const float* pos_w1    = (const float*)d_in[3];
  const float* pos_b1    = (const float*)d_in[4];
  const float* pos_w2    = (const float*)d_in[5];
  const float* pos_b2    = (const float*)d_in[6];
  const float* rms_w     = (const float*)d_in[7];
  const float* in_proj_w = (const float*)d_in[8];
  const float* conv_w    = (const float*)d_in[9];
  const float* conv_b    = (const float*)d_in[10];
  const float* x_proj_w  = (const float*)d_in[11];
  const float* dt_proj_w = (const float*)d_in[12];
  const float* dt_proj_b = (const float*)d_in[13];
  const float* A_log     = (const float*)d_in[14];
  const float* Dp        = (const float*)d_in[15];
  const float* out_proj_w= (const float*)d_in[16];
  const float* ssm_w     = (const float*)d_in[17];
  const float* ssm_b     = (const float*)d_in[18];

  float* ws   = (float*)d_ws;
  const size_t N = NTOK;
  float* pos  = ws;                   // N*5
  float* fbuf = pos  + N * 5;         // N*128 (also reused for out_proj result)
  float* hbuf = fbuf + N * DIMF;      // N*128
  float* xz   = hbuf + N * DIMF;      // N*512
  float* xcb  = xz   + N * 512;       // N*256
  float* dbl  = xcb  + N * DIN;       // N*40
  float* dtb  = dbl  + N * 40;        // N*256
  float* yyb  = dtb  + N * DIN;       // N*256
  int*   ranks = (int*)(yyb + N * DIN);
  int*   perm  = ranks + N;
  int*   invp  = perm + N;
  float* obuf  = fbuf;                // reuse: f consumed before out_proj

  k_pos<<<NTOK / 256, 256, 0, stream>>>(coords, curve, pos, ranks);
  k_pe<<<NTOK / 64, 256, 0, stream>>>(pos, pos_w1, pos_b1, pos_w2, pos_b2,
                                      feats, fbuf);
  k_sort<<<NB, 1024, 0, stream>>>(ranks, perm, invp);
  k_rms_perm<<<NTOK, 128, 0, stream>>>(fbuf, perm, rms_w, hbuf);
  k_gemm<128, 512, 512><<<NTOK / 64, 256, 0, stream>>>(hbuf, in_proj_w, xz);
  k_conv<<<NTOK, 256, 0, stream>>>(xz, conv_w, conv_b, xcb);
  k_gemm<256, 48, 40><<<NTOK / 64, 256, 0, stream>>>(xcb, x_proj_w, dbl);
  k_dt<<<NTOK, 256, 0, stream>>>(dbl, dt_proj_w, dt_proj_b, dtb);
  k_scan<<<NB, 256, 0, stream>>>(dtb, xcb, dbl, xz, A_log, Dp, yyb);
  k_gemm<256, 128, 128><<<NTOK / 64, 256, 0, stream>>>(yyb, out_proj_w, obuf);
  k_ln<<<NTOK, 128, 0, stream>>>(obuf, invp, ssm_w, ssm_b, (float*)d_out);
}